// GraphSage_73564199845998
// MI455X (gfx1250) — compile-verified
//
#include <hip/hip_runtime.h>
#include <hip/hip_bf16.h>
#include <math.h>

typedef __attribute__((ext_vector_type(2))) float v2f;
typedef __attribute__((ext_vector_type(8))) float v8f;

#define N_NODES 50000
#define DIMS 128
#define ROW_TILES (N_NODES / 16)   // 3125, exact (no row tail)
#define WAVES_PER_BLOCK 8

// ---------------------------------------------------------------------------
// Kernel 1: Y = relu(X @ W^T + b)   [N,128] x [128,128]
// One wave computes a 16-row x 128-col tile via 8 accumulator tiles.
// W (64 KB) staged in LDS; B-fragments read from LDS, A from global (L2-hot).
// ---------------------------------------------------------------------------
__global__ __launch_bounds__(256) void k_lin_relu(
    const float* __restrict__ X, const float* __restrict__ W,
    const float* __restrict__ bias, float* __restrict__ Y, int rowTiles) {
  __shared__ float sW[DIMS * DIMS];  // 64 KB: W[e][k] row-major

  const int tid = threadIdx.x;
  {
    const float4* Wv = (const float4*)W;
    float4* sWv = (float4*)sW;
#pragma unroll
    for (int i = 0; i < (DIMS * DIMS / 4) / 256; ++i)
      sWv[tid + i * 256] = Wv[tid + i * 256];
  }
  __syncthreads();

  const int lane = tid & 31;
  const int wave = tid >> 5;
  const int rowTile = blockIdx.x * WAVES_PER_BLOCK + wave;
  if (rowTile >= rowTiles) return;  // wave-uniform: EXEC all-ones inside

  const int row0 = rowTile * 16;
  const int m = lane & 15;              // M (A) / N (B) index within tile
  const int khi = (lane >> 4) << 1;     // lanes 16-31 hold K=k0+2,k0+3

  v8f acc[8] = {};
  const float* Arow = X + (size_t)(row0 + m) * DIMS;

#pragma unroll 4
  for (int k0 = 0; k0 < DIMS; k0 += 4) {
    const int kk = k0 + khi;
    v2f a;
    a.x = Arow[kk];
    a.y = Arow[kk + 1];
#pragma unroll
    for (int j = 0; j < 8; ++j) {
      const int n = j * 16 + m;
      v2f b;                     // B[k][n] = W[n][k]
      b.x = sW[n * DIMS + kk];
      b.y = sW[n * DIMS + kk + 1];
      acc[j] = __builtin_amdgcn_wmma_f32_16x16x4_f32(
          false, a, false, b, (short)0, acc[j], false, false);
    }
  }

  // Epilogue: + bias, relu, store. C/D layout: VGPR i, lanes<16 -> M=i,
  // lanes>=16 -> M=i+8; N = lane&15.
  const int rowA = row0 + ((lane >> 4) << 3);
#pragma unroll
  for (int j = 0; j < 8; ++j) {
    const int col = j * 16 + m;
    const float bi = bias[col];
#pragma unroll
    for (int i = 0; i < 8; ++i) {
      float v = acc[j][i] + bi;
      v = v > 0.f ? v : 0.f;
      Y[(size_t)(rowA + i) * DIMS + col] = v;
    }
  }
}

// ---------------------------------------------------------------------------
// Kernel 2: agg[n] = max_s Y[neigh[n][s]]   (gather-max, L2-resident table)
// One wave per node; each lane owns 4 consecutive columns (float4).
// S is a template constant so the loop fully unrolls: all index loads become
// hoisted scalar loads and up to S independent global_load_b128 stay in
// flight per wave (this stage is L2-latency bound, so MLP is everything).
// ---------------------------------------------------------------------------
template <int S>
__global__ __launch_bounds__(256) void k_gather_max(
    const float* __restrict__ Y, const int* __restrict__ neigh,
    float* __restrict__ agg) {
  const int lane = threadIdx.x & 31;
  const int wave = threadIdx.x >> 5;
  const int n = blockIdx.x * WAVES_PER_BLOCK + wave;
  if (n >= N_NODES) return;

  const int* nb = neigh + (size_t)n * S;
  // Load all wave-uniform indices up front (scalarizable).
  int idx[S];
#pragma unroll
  for (int s = 0; s < S; ++s) idx[s] = nb[s];

  float4 acc = make_float4(-INFINITY, -INFINITY, -INFINITY, -INFINITY);
#pragma unroll
  for (int s = 0; s < S; ++s) {
    const float4 v = *(const float4*)(Y + (size_t)idx[s] * DIMS + lane * 4);
    acc.x = fmaxf(acc.x, v.x);
    acc.y = fmaxf(acc.y, v.y);
    acc.z = fmaxf(acc.z, v.z);
    acc.w = fmaxf(acc.w, v.w);
  }
  *(float4*)(agg + (size_t)n * DIMS + lane * 4) = acc;
}

// ---------------------------------------------------------------------------
// Kernel 3: out = l2norm_rows(relu([E, A] @ Wu^T))
// K=256 split into two 128-K phases (E then A); Wu half staged in LDS per
// phase. Row L2-norm fused via shuffle reduction within 16-lane halves.
// ---------------------------------------------------------------------------
__global__ __launch_bounds__(256) void k_update_norm(
    const float* __restrict__ E, const float* __restrict__ A,
    const float* __restrict__ Wu,  // [128][256] row-major
    float* __restrict__ out, int rowTiles) {
  __shared__ float sW[DIMS * DIMS];  // 64 KB per phase

  const int tid = threadIdx.x;
  const int lane = tid & 31;
  const int wave = tid >> 5;
  const int rowTile = blockIdx.x * WAVES_PER_BLOCK + wave;
  const int row0 = rowTile * 16;
  const int m = lane & 15;
  const int khi = (lane >> 4) << 1;
  const bool active = rowTile < rowTiles;  // wave-uniform

  v8f acc[8] = {};

  for (int phase = 0; phase < 2; ++phase) {
    __syncthreads();
    // stage Wu[:, phase*128 : phase*128+128] as sW[n][k]
    for (int i = tid; i < DIMS * DIMS / 2; i += 256) {
      const int nrow = i / 64;           // (i*2)/128
      const int kcol = (i * 2) & (DIMS - 1);
      *(float2*)&sW[nrow * DIMS + kcol] =
          *(const float2*)&Wu[(size_t)nrow * (2 * DIMS) + phase * DIMS + kcol];
    }
    __syncthreads();
    if (!active) continue;

    const float* X = phase ? A : E;
    const float* Arow = X + (size_t)(row0 + m) * DIMS;
#pragma unroll 4
    for (int k0 = 0; k0 < DIMS; k0 += 4) {
      const int kk = k0 + khi;
      v2f a;
      a.x = Arow[kk];
      a.y = Arow[kk + 1];
#pragma unroll
      for (int j = 0; j < 8; ++j) {
        const int n = j * 16 + m;
        v2f b;
        b.x = sW[n * DIMS + kk];
        b.y = sW[n * DIMS + kk + 1];
        acc[j] = __builtin_amdgcn_wmma_f32_16x16x4_f32(
            false, a, false, b, (short)0, acc[j], false, false);
      }
    }
  }
  if (!active) return;

  // relu + per-row sum of squares. Row (i + 8*(lane>=16)) lives in one
  // 16-lane half; reduce across the half with xor masks 1..8.
  float ss[8];
#pragma unroll
  for (int i = 0; i < 8; ++i) ss[i] = 0.f;
#pragma unroll
  for (int j = 0; j < 8; ++j)
#pragma unroll
    for (int i = 0; i < 8; ++i) {
      float v = fmaxf(acc[j][i], 0.f);
      acc[j][i] = v;
      ss[i] += v * v;
    }
#pragma unroll
  for (int msk = 1; msk <= 8; msk <<= 1)
#pragma unroll
    for (int i = 0; i < 8; ++i) ss[i] += __shfl_xor(ss[i], msk, 32);

  float scale[8];
#pragma unroll
  for (int i = 0; i < 8; ++i)
    scale[i] = 1.f / fmaxf(sqrtf(ss[i]), 1e-12f);

  const int rowA = row0 + ((lane >> 4) << 3);
#pragma unroll
  for (int j = 0; j < 8; ++j) {
    const int col = j * 16 + m;
#pragma unroll
    for (int i = 0; i < 8; ++i)
      out[(size_t)(rowA + i) * DIMS + col] = acc[j][i] * scale[i];
  }
}

// ---------------------------------------------------------------------------
extern "C" void kernel_launch(void* const* d_in, const int* in_sizes, int n_in,
                              void* d_out, int out_size, void* d_ws,
                              size_t ws_size, hipStream_t stream) {
  const float* X0   = (const float*)d_in[0];  // node_features [N,128]
  const float* Wagg = (const float*)d_in[1];  // [128,128]
  const float* bagg = (const float*)d_in[2];  // [128]
  const float* Wupd = (const float*)d_in[3];  // [128,256]
  const int* n1     = (const int*)d_in[4];    // [N,25]
  const int* n2     = (const int*)d_in[5];    // [N,10]
  float* out = (float*)d_out;

  float* Y    = (float*)d_ws;                      // [N,128]
  float* agg  = Y + (size_t)N_NODES * DIMS;        // [N,128]
  float* emb1 = agg + (size_t)N_NODES * DIMS;      // [N,128]

  const dim3 blk(256);
  const int gemmGrid = (ROW_TILES + WAVES_PER_BLOCK - 1) / WAVES_PER_BLOCK;
  const int nodeGrid = (N_NODES + WAVES_PER_BLOCK - 1) / WAVES_PER_BLOCK;

  // Layer 1 (hoisted aggregator linear): Y = relu(X0@Wagg^T+b); agg = max
  // over neigh1 of Y; emb1 = norm(relu([X0,agg]@Wupd^T))
  k_lin_relu<<<gemmGrid, blk, 0, stream>>>(X0, Wagg, bagg, Y, ROW_TILES);
  k_gather_max<25><<<nodeGrid, blk, 0, stream>>>(Y, n1, agg);
  k_update_norm<<<gemmGrid, blk, 0, stream>>>(X0, agg, Wupd, emb1, ROW_TILES);

  // Layer 2
  k_lin_relu<<<gemmGrid, blk, 0, stream>>>(emb1, Wagg, bagg, Y, ROW_TILES);
  k_gather_max<10><<<nodeGrid, blk, 0, stream>>>(Y, n2, agg);
  k_update_norm<<<gemmGrid, blk, 0, stream>>>(emb1, agg, Wupd, out, ROW_TILES);
}